// ClusterGCN_79955111182426
// MI455X (gfx1250) — compile-verified
//
#include <hip/hip_runtime.h>
#include <hip/hip_bf16.h>

typedef __attribute__((ext_vector_type(2))) float v2f;
typedef __attribute__((ext_vector_type(8))) float v8f;

#define N_NODES 100000
#define C_IN    128
#define M_TILES (N_NODES / 16)   // 6250, exact

// ---------------------------------------------------------------------------
// Degree: one thread per edge, count non-self edges at their target (col).
// ---------------------------------------------------------------------------
__global__ void deg_kernel(const int* __restrict__ row, const int* __restrict__ col,
                           float* __restrict__ deg, int E) {
    int e = blockIdx.x * blockDim.x + threadIdx.x;
    if (e >= E) return;
    int r = row[e], c = col[e];
    if (r != c) atomicAdd(&deg[c], 1.0f);
}

// ---------------------------------------------------------------------------
// Scatter: one wave per edge. 32 lanes x float4 covers all 128 channels, so
// the gather of X[row] is a fully coalesced 512B burst (hits the 192MB L2).
// Scatter-add via global_atomic_add_f32 into agg[col].
// ---------------------------------------------------------------------------
__global__ void scatter_kernel(const float* __restrict__ X,
                               const int* __restrict__ row, const int* __restrict__ col,
                               float* __restrict__ agg, int E) {
    long long tid = (long long)blockIdx.x * blockDim.x + threadIdx.x;
    int e = (int)(tid >> 5);
    if (e >= E) return;
    int c4 = ((int)tid & 31) * 4;
    int r = row[e], c = col[e];
    if (r == c) return;
    const float4 v = *(const float4*)(X + (size_t)r * C_IN + c4);
    float* dst = agg + (size_t)c * C_IN + c4;
    atomicAdd(dst + 0, v.x);
    atomicAdd(dst + 1, v.y);
    atomicAdd(dst + 2, v.z);
    atomicAdd(dst + 3, v.w);
}

// ---------------------------------------------------------------------------
// Fused ClusterGCN layer GEMM, V_WMMA_F32_16X16X4_F32 (exact fp32), with
// register blocking over the N dimension: one wave computes one 16-row
// m-tile for ALL NT n-tiles, so A fragments are loaded once per k-step and
// reused NT times (A L2 traffic drops 8x for layer 1, 4x for layer 2).
//   Out[m,n] = sum_k ((Agg[m,k]+Self[m,k]) * dinv[m]) * Wout[n,k]
//            + sum_k Self[m,k] * Wroot[n,k] + bias[n]     (optional ReLU)
// A-fragment (16x4 f32): lane L<16 holds row M=L, K={k0,k0+1}; L>=16 holds
// row M=L-16, K={k0+2,k0+3}. B-fragment mirrored with lanes = N columns.
// C/D: VGPR g, lanes 0-15 -> M=g, lanes 16-31 -> M=g+8, N = lane%16.
// ---------------------------------------------------------------------------
template <int RELU, int NT>
__global__ void gcn_gemm_kernel(const float* __restrict__ Agg,   // [N,128] neighbor sums
                                const float* __restrict__ Self,  // [N,128] self features
                                const float* __restrict__ deg,   // [N] neighbor count
                                const float* __restrict__ Wout,  // [NT*16,128] row-major
                                const float* __restrict__ bias,  // [NT*16]
                                const float* __restrict__ Wroot, // [NT*16,128]
                                float* __restrict__ Out) {       // [N,NT*16]
    const int lane  = threadIdx.x & 31;
    const int wave  = threadIdx.x >> 5;
    const int mTile = blockIdx.x * (blockDim.x >> 5) + wave;
    if (mTile >= M_TILES) return;           // wave-uniform; EXEC stays all-1s

    const int half = lane >> 4;             // selects K sub-pair
    const int mr   = lane & 15;
    const int m    = mTile * 16 + mr;       // A row for this lane (both halves)
    const int Cout = NT * 16;

    const float dinv = 1.0f / (deg[m] + 1.0f);
    const float* aggRow  = Agg   + (size_t)m * C_IN;
    const float* selfRow = Self  + (size_t)m * C_IN;
    // B column for this lane is n = nt*16 + mr; nt handled via immediate offset.
    const float* woBase  = Wout  + (size_t)mr * C_IN;
    const float* wrBase  = Wroot + (size_t)mr * C_IN;

    v8f acc[NT] = {};

    for (int k0 = 0; k0 < C_IN; k0 += 4) {
        const int ka = k0 + 2 * half;
        v2f as = *(const v2f*)(aggRow  + ka);
        v2f ax = *(const v2f*)(selfRow + ka);
        v2f a;
        a.x = (as.x + ax.x) * dinv;
        a.y = (as.y + ax.y) * dinv;
#pragma unroll
        for (int nt = 0; nt < NT; ++nt) {
            v2f bo = *(const v2f*)(woBase + nt * 16 * C_IN + ka);
            v2f br = *(const v2f*)(wrBase + nt * 16 * C_IN + ka);
            acc[nt] = __builtin_amdgcn_wmma_f32_16x16x4_f32(false, a,  false, bo,
                                                            (short)0, acc[nt], false, false);
            acc[nt] = __builtin_amdgcn_wmma_f32_16x16x4_f32(false, ax, false, br,
                                                            (short)0, acc[nt], false, false);
        }
    }

#pragma unroll
    for (int nt = 0; nt < NT; ++nt) {
        const int   outCol = nt * 16 + mr;
        const float b      = bias[outCol];
#pragma unroll
        for (int g = 0; g < 8; ++g) {
            const int outRow = mTile * 16 + g + 8 * half;
            float v = acc[nt][g] + b;
            if (RELU) v = fmaxf(v, 0.0f);
            Out[(size_t)outRow * Cout + outCol] = v;
        }
    }
}

// ---------------------------------------------------------------------------
// Launcher: deg -> scatter1 -> GEMM1(+ReLU) -> scatter2 -> GEMM2 -> d_out
// ---------------------------------------------------------------------------
extern "C" void kernel_launch(void* const* d_in, const int* in_sizes, int n_in,
                              void* d_out, int out_size, void* d_ws, size_t ws_size,
                              hipStream_t stream) {
    const float* x   = (const float*)d_in[0];
    const int*   ei  = (const int*)  d_in[1];
    const float* W1o = (const float*)d_in[2];
    const float* b1  = (const float*)d_in[3];
    const float* W1r = (const float*)d_in[4];
    const float* W2o = (const float*)d_in[5];
    const float* b2  = (const float*)d_in[6];
    const float* W2r = (const float*)d_in[7];
    float*       out = (float*)d_out;

    const int E = in_sizes[1] / 2;
    const int* row = ei;
    const int* col = ei + E;

    // Workspace layout (floats): deg[N] | agg[N*128] | h[N*128]
    float* deg = (float*)d_ws;
    float* agg = deg + ((N_NODES + 63) & ~63);
    float* h   = agg + (size_t)N_NODES * C_IN;

    const size_t aggBytes = (size_t)N_NODES * C_IN * sizeof(float);

    hipMemsetAsync(deg, 0, N_NODES * sizeof(float), stream);
    hipMemsetAsync(agg, 0, aggBytes, stream);

    deg_kernel<<<(E + 255) / 256, 256, 0, stream>>>(row, col, deg, E);

    const long long sthreads = (long long)E * 32;
    const int sblocks = (int)((sthreads + 255) / 256);

    const int wavesPerBlock = 8;                       // 256 threads
    const int gemmBlocks = (M_TILES + wavesPerBlock - 1) / wavesPerBlock;

    // ---- Layer 1: Cin=128 -> Chid=128 (NT=8 n-tiles per wave) ----
    scatter_kernel<<<sblocks, 256, 0, stream>>>(x, row, col, agg, E);
    gcn_gemm_kernel<1, 8><<<gemmBlocks, 256, 0, stream>>>(
        agg, x, deg, W1o, b1, W1r, h);

    // ---- Layer 2: Chid=128 -> Cout=64 (NT=4 n-tiles per wave) ----
    hipMemsetAsync(agg, 0, aggBytes, stream);
    scatter_kernel<<<sblocks, 256, 0, stream>>>(h, row, col, agg, E);
    gcn_gemm_kernel<0, 4><<<gemmBlocks, 256, 0, stream>>>(
        agg, h, deg, W2o, b2, W2r, out);
}